// SpGraphTransAttentionLayer_12266426597865
// MI455X (gfx1250) — compile-verified
//
#include <hip/hip_runtime.h>
#include <math.h>

#define B_    4
#define N_    10000
#define E_    160000
#define FIN   64
#define DATT  128
#define H_    8
#define BN    (B_ * N_)          // 40000 rows (16 | 40000)
#define MT    (BN / 16)          // 2500 row tiles
#define NT    (DATT / 16)        // 8 col tiles
#define TILES (MT * NT * 3)      // 60000 tiles, 8 waves/block -> 7500 blocks

typedef float v2f __attribute__((ext_vector_type(2)));
typedef float v8f __attribute__((ext_vector_type(8)));

// ---------------- stats init ----------------
__global__ void init_stats(float* __restrict__ maxbuf, float* __restrict__ sumbuf) {
    int i = threadIdx.x;
    if (i < B_ * H_) { maxbuf[i] = -INFINITY; sumbuf[i] = 0.0f; }
}

// ---------------- fused QKV projection via V_WMMA_F32_16X16X4_F32 ----------------
// Y(40000x128) = X(40000x64) @ W^T(64x128) + b, for W in {Qw,Kw,Vw}.
// One wave computes one 16x16 tile; K swept in steps of 4 (16 WMMA ops/tile).
__global__ void __launch_bounds__(256) qkv_proj_wmma(
    const float* __restrict__ x,
    const float* __restrict__ Qw, const float* __restrict__ Qb,
    const float* __restrict__ Kw, const float* __restrict__ Kb,
    const float* __restrict__ Vw, const float* __restrict__ Vb,
    float* __restrict__ qbuf, float* __restrict__ kbuf,
    float* __restrict__ vout)
{
    const int lane = threadIdx.x & 31;
    const int wave = threadIdx.x >> 5;
    const int tile = blockIdx.x * (blockDim.x >> 5) + wave;
    if (tile >= TILES) return;              // wave-uniform: EXEC stays all-ones
    const int sub = tile % (NT * 3);        // 0..23
    const int m   = tile / (NT * 3);        // row tile 0..2499
    const int w   = sub >> 3;               // 0=Q 1=K 2=V
    const int nt  = sub & 7;                // col tile 0..7

    const float* Wm; const float* bias;
    if (w == 0)      { Wm = Qw; bias = Qb; }
    else if (w == 1) { Wm = Kw; bias = Kb; }
    else             { Wm = Vw; bias = Vb; }

    const int half = lane >> 4;             // 0: K={0,1}, 1: K={2,3} (per ISA 16x4 layout)
    const int l16  = lane & 15;
    const int arow = m * 16 + l16;          // A-matrix M index for this lane
    const int ncol = nt * 16 + l16;         // B-matrix N index (column of D_ATT)
    const int kh   = half * 2;

    const float* ap = x  + (size_t)arow * FIN + kh;   // X row-major, stride 64
    const float* bp = Wm + (size_t)ncol * FIN + kh;   // W (128,64): W^T[k][n] = W[n*64+k]

    v8f c = {};
#pragma unroll
    for (int kk = 0; kk < FIN; kk += 4) {
        v2f a = *(const v2f*)(ap + kk);     // K = kk+kh, kk+kh+1
        v2f b = *(const v2f*)(bp + kk);
        c = __builtin_amdgcn_wmma_f32_16x16x4_f32(
                /*neg_a=*/false, a, /*neg_b=*/false, b,
                /*c_mod=*/(short)0, c, /*reuse_a=*/false, /*reuse_b=*/false);
    }

    const float bv = bias[ncol];
    // D layout: VGPR r -> M = r + 8*half, N = l16  (per 16x16 f32 C/D layout)
    const int rbase = m * 16 + half * 8;
    if (w == 0) {
#pragma unroll
        for (int r = 0; r < 8; ++r)
            qbuf[(size_t)(rbase + r) * DATT + ncol] = c[r] + bv;
    } else if (w == 1) {
#pragma unroll
        for (int r = 0; r < 8; ++r)
            kbuf[(size_t)(rbase + r) * DATT + ncol] = c[r] + bv;
    } else {
        // v output layout (b,n,dk,h): a = h*16+dk -> offset dk*8 + h within row
        const int voff = (ncol & 15) * H_ + (ncol >> 4);
#pragma unroll
        for (int r = 0; r < 8; ++r)
            vout[(size_t)(rbase + r) * DATT + voff] = c[r] + bv;
    }
}

// ---------------- float atomic max via ordered-int trick ----------------
__device__ __forceinline__ void atomicMaxFloat(float* addr, float val) {
    if (val >= 0.0f)
        atomicMax((int*)addr, __float_as_int(val));
    else
        atomicMin((unsigned int*)addr, __float_as_uint(val));
}

// ---------------- pass 1: edge scores + per-(b,h) running max ----------------
__global__ void __launch_bounds__(256) edge_scores(
    const float* __restrict__ qbuf, const float* __restrict__ kbuf,
    const long long* __restrict__ edge, float* __restrict__ attn,
    float* __restrict__ maxbuf)
{
    __shared__ float red[H_][256];
    const int tid = threadIdx.x;
    const int e   = blockIdx.x * 256 + tid;
    const int b   = blockIdx.y;

    float p[H_];
    if (e < E_) {
        const long long s = edge[e];        // edge[0][e] -> q
        const long long d = edge[E_ + e];   // edge[1][e] -> k
        const float4* qr = (const float4*)(qbuf + ((size_t)b * N_ + (size_t)s) * DATT);
        const float4* kr = (const float4*)(kbuf + ((size_t)b * N_ + (size_t)d) * DATT);
#pragma unroll
        for (int h = 0; h < H_; ++h) {
            float acc = 0.0f;
#pragma unroll
            for (int j = 0; j < 4; ++j) {
                float4 a = qr[h * 4 + j];
                float4 k = kr[h * 4 + j];
                acc += a.x * k.x + a.y * k.y + a.z * k.z + a.w * k.w;
            }
            p[h] = acc * 0.25f;             // 1/sqrt(D_K), D_K=16
        }
        float4* o = (float4*)(attn + ((size_t)b * E_ + e) * H_);
        o[0] = make_float4(p[0], p[1], p[2], p[3]);
        o[1] = make_float4(p[4], p[5], p[6], p[7]);
    } else {
#pragma unroll
        for (int h = 0; h < H_; ++h) p[h] = -INFINITY;
    }

#pragma unroll
    for (int h = 0; h < H_; ++h) red[h][tid] = p[h];
    __syncthreads();
    for (int s = 128; s > 0; s >>= 1) {
        if (tid < s) {
#pragma unroll
            for (int h = 0; h < H_; ++h)
                red[h][tid] = fmaxf(red[h][tid], red[h][tid + s]);
        }
        __syncthreads();
    }
    if (tid == 0) {
#pragma unroll
        for (int h = 0; h < H_; ++h)
            atomicMaxFloat(&maxbuf[b * H_ + h], red[h][0]);
    }
}

// ---------------- pass 2: exp(p - max) + per-(b,h) sum ----------------
__global__ void __launch_bounds__(256) edge_exp_sum(
    float* __restrict__ attn, const float* __restrict__ maxbuf,
    float* __restrict__ sumbuf)
{
    __shared__ float red[H_][256];
    const int tid = threadIdx.x;
    const int e   = blockIdx.x * 256 + tid;
    const int b   = blockIdx.y;

    float mv[H_];
#pragma unroll
    for (int h = 0; h < H_; ++h) mv[h] = maxbuf[b * H_ + h];

    float p[H_];
#pragma unroll
    for (int h = 0; h < H_; ++h) p[h] = 0.0f;

    if (e < E_) {
        float4* o = (float4*)(attn + ((size_t)b * E_ + e) * H_);
        float4 a0 = o[0], a1 = o[1];
        p[0] = __expf(a0.x - mv[0]); p[1] = __expf(a0.y - mv[1]);
        p[2] = __expf(a0.z - mv[2]); p[3] = __expf(a0.w - mv[3]);
        p[4] = __expf(a1.x - mv[4]); p[5] = __expf(a1.y - mv[5]);
        p[6] = __expf(a1.z - mv[6]); p[7] = __expf(a1.w - mv[7]);
        o[0] = make_float4(p[0], p[1], p[2], p[3]);
        o[1] = make_float4(p[4], p[5], p[6], p[7]);
    }

#pragma unroll
    for (int h = 0; h < H_; ++h) red[h][tid] = p[h];
    __syncthreads();
    for (int s = 128; s > 0; s >>= 1) {
        if (tid < s) {
#pragma unroll
            for (int h = 0; h < H_; ++h)
                red[h][tid] += red[h][tid + s];
        }
        __syncthreads();
    }
    if (tid == 0) {
#pragma unroll
        for (int h = 0; h < H_; ++h)
            atomicAdd(&sumbuf[b * H_ + h], red[h][0]);
    }
}

// ---------------- pass 3: normalize ----------------
__global__ void __launch_bounds__(256) edge_norm(
    float* __restrict__ attn, const float* __restrict__ sumbuf)
{
    const int e = blockIdx.x * 256 + threadIdx.x;
    const int b = blockIdx.y;
    if (e >= E_) return;
    float inv[H_];
#pragma unroll
    for (int h = 0; h < H_; ++h) inv[h] = 1.0f / sumbuf[b * H_ + h];
    float4* o = (float4*)(attn + ((size_t)b * E_ + e) * H_);
    float4 a0 = o[0], a1 = o[1];
    a0.x *= inv[0]; a0.y *= inv[1]; a0.z *= inv[2]; a0.w *= inv[3];
    a1.x *= inv[4]; a1.y *= inv[5]; a1.z *= inv[6]; a1.w *= inv[7];
    o[0] = a0; o[1] = a1;
}

extern "C" void kernel_launch(void* const* d_in, const int* in_sizes, int n_in,
                              void* d_out, int out_size, void* d_ws, size_t ws_size,
                              hipStream_t stream)
{
    const float*     x    = (const float*)d_in[0];
    const float*     Qw   = (const float*)d_in[1];
    const float*     Qb   = (const float*)d_in[2];
    const float*     Kw   = (const float*)d_in[3];
    const float*     Kb   = (const float*)d_in[4];
    const float*     Vw   = (const float*)d_in[5];
    const float*     Vb   = (const float*)d_in[6];
    const long long* edge = (const long long*)d_in[7];

    float* attn = (float*)d_out;                          // (B,E,H)
    float* vout = attn + (size_t)B_ * E_ * H_;            // (B,N,D_K,H)

    float* qbuf   = (float*)d_ws;                         // (BN,128)
    float* kbuf   = qbuf + (size_t)BN * DATT;             // (BN,128)
    float* maxbuf = kbuf + (size_t)BN * DATT;             // 32 floats
    float* sumbuf = maxbuf + B_ * H_;                     // 32 floats

    init_stats<<<1, 64, 0, stream>>>(maxbuf, sumbuf);

    qkv_proj_wmma<<<TILES / 8, 256, 0, stream>>>(
        x, Qw, Qb, Kw, Kb, Vw, Vb, qbuf, kbuf, vout);

    dim3 eg(E_ / 256, B_);   // 625 x 4, E_ divisible by 256
    edge_scores <<<eg, 256, 0, stream>>>(qbuf, kbuf, edge, attn, maxbuf);
    edge_exp_sum<<<eg, 256, 0, stream>>>(attn, maxbuf, sumbuf);
    edge_norm   <<<eg, 256, 0, stream>>>(attn, sumbuf);
}